// _EquivariantLocalMessageDenoiser_17420387353217
// MI455X (gfx1250) — compile-verified
//
#include <hip/hip_runtime.h>

typedef __attribute__((ext_vector_type(2))) float v2f;
typedef __attribute__((ext_vector_type(8))) float v8f;

#define N_NODES 1024
#define HDIM    256
#define EHDIM   128

__device__ __forceinline__ float silu_f(float x) { return x / (1.0f + __expf(-x)); }

// One wave per 16x16 output tile: out = silu(feat @ W + b), fp32 WMMA 16x16x4.
// feat = [s0 (HDIM wide) | t1 (3) | t2 (3)] when TAIL, else just s0 (HDIM wide).
// A frag (16x4 f32): lanes 0-15 hold K=0,1; lanes 16-31 hold K=2,3 (M = lane%16).
// B frag (4x16 f32): VGPR0 = rows K=0 / K=2 per half, VGPR1 = rows K=1 / K=3.
// D (16x16 f32): VGPR r -> M = r + 8*half, N = lane%16.
template <bool TAIL>
__global__ void gemm_bias_silu_wmma(const float* __restrict__ s0,
                                    const float* __restrict__ t1,
                                    const float* __restrict__ t2,
                                    const float* __restrict__ W,
                                    const float* __restrict__ bias,
                                    float* __restrict__ out,
                                    int Nout) {
  const int lane = threadIdx.x;
  const int half = lane >> 4;
  const int li   = lane & 15;
  const int c0   = blockIdx.x * 16;
  const int m0   = blockIdx.y * 16;
  const int col  = c0 + li;
  const int row  = m0 + li;

  // Branch-free hot loop over the aligned HDIM-wide segment.
  const float* __restrict__ arow = s0 + (size_t)row * HDIM + (half << 1);
  const float* __restrict__ wcol = W + col;

  v8f acc = {0.f,0.f,0.f,0.f,0.f,0.f,0.f,0.f};
  #pragma unroll 4
  for (int s = 0; s < HDIM / 4; ++s) {
    const int k = (s << 2) + (half << 1);
    const v2f a = *(const v2f*)(arow + (s << 2));
    v2f b;
    b.x = wcol[(size_t)k * Nout];
    b.y = wcol[(size_t)(k + 1) * Nout];
    acc = __builtin_amdgcn_wmma_f32_16x16x4_f32(false, a, false, b, (short)0, acc, false, false);
  }

  if (TAIL) {
    // 6 extra features: k = 256..258 from t1, 259..261 from t2, 262..263 zero.
    const float f0 = t1[row * 3 + 0], f1 = t1[row * 3 + 1], f2 = t1[row * 3 + 2];
    const float f3 = t2[row * 3 + 0], f4 = t2[row * 3 + 1], f5 = t2[row * 3 + 2];
    v2f a, b;
    // step k0=256: half0 -> k=256,257 ; half1 -> k=258,259
    a.x = half ? f2 : f0;
    a.y = half ? f3 : f1;
    b.x = half ? wcol[(size_t)258 * Nout] : wcol[(size_t)256 * Nout];
    b.y = half ? wcol[(size_t)259 * Nout] : wcol[(size_t)257 * Nout];
    acc = __builtin_amdgcn_wmma_f32_16x16x4_f32(false, a, false, b, (short)0, acc, false, false);
    // step k0=260: half0 -> k=260,261 ; half1 -> k=262,263 (zero pad)
    a.x = half ? 0.f : f4;
    a.y = half ? 0.f : f5;
    b.x = half ? 0.f : wcol[(size_t)260 * Nout];
    b.y = half ? 0.f : wcol[(size_t)261 * Nout];
    acc = __builtin_amdgcn_wmma_f32_16x16x4_f32(false, a, false, b, (short)0, acc, false, false);
  }

  const float bc = bias[col];
  #pragma unroll
  for (int r = 0; r < 8; ++r) {
    const int orow = m0 + r + (half << 3);
    out[(size_t)orow * Nout + col] = silu_f(acc[r] + bc);
  }
}

// dst = nh @ dst_w + dst_b ; ns = sigmoid(nh @ ns_w + ns_b). One wave per node.
// (src head is skipped: src[n] is constant along the softmax axis and cancels.)
__global__ void heads_kernel(const float* __restrict__ nh,
                             const float* __restrict__ dst_w, const float* __restrict__ dst_b,
                             const float* __restrict__ ns_w,  const float* __restrict__ ns_b,
                             float* __restrict__ dstv, float* __restrict__ nsv) {
  const int wave = threadIdx.x >> 5;
  const int lane = threadIdx.x & 31;
  const int node = blockIdx.x * 8 + wave;
  float pd = 0.f, pn = 0.f;
  #pragma unroll
  for (int j = 0; j < 8; ++j) {
    const int k = lane + (j << 5);
    const float v = nh[node * HDIM + k];
    pd += v * dst_w[k];
    pn += v * ns_w[k];
  }
  #pragma unroll
  for (int m = 16; m; m >>= 1) {
    pd += __shfl_xor(pd, m, 32);
    pn += __shfl_xor(pn, m, 32);
  }
  if (lane == 0) {
    dstv[node] = pd + dst_b[0];
    nsv[node]  = 1.0f / (1.0f + __expf(-(pn + ns_b[0])));
  }
}

// One block per row n. Edge MLP (4->128->1) on the WMMA f32 pipe, then
// softmax over m (diag masked, renormalized) and local_messages in LDS.
// eg1/eg2 tables are staged in LDS (block-wide constants) to keep VGPR
// pressure low; the g-loop is only 2-deep pipelined so no spills occur.
__global__ void edge_attn_msg_kernel(const float* __restrict__ x,
                                     const float* __restrict__ frames,
                                     const float* __restrict__ dstv,
                                     const float* __restrict__ nsv,
                                     const float* __restrict__ eg1_w,
                                     const float* __restrict__ eg1_b,
                                     const float* __restrict__ eg2_w,
                                     float* __restrict__ msg) {
  __shared__ float s_logit[N_NODES];
  __shared__ float s_rl0[N_NODES];
  __shared__ float s_rl1[N_NODES];
  __shared__ float s_rl2[N_NODES];
  __shared__ float s_w1[4 * EHDIM];   // eg1_w (4 x 128)
  __shared__ float s_b1[EHDIM];       // eg1_b
  __shared__ float s_w2[EHDIM];       // eg2_w
  __shared__ float s_red[32];

  const int n    = blockIdx.x;
  const int tid  = threadIdx.x;
  const int wave = tid >> 5;
  const int lane = tid & 31;
  const int half = lane >> 4;
  const int li   = lane & 15;

  // Stage edge-MLP weights in LDS.
  #pragma unroll
  for (int i = tid; i < 4 * EHDIM; i += 256) s_w1[i] = eg1_w[i];
  if (tid < EHDIM) { s_b1[tid] = eg1_b[tid]; s_w2[tid] = eg2_w[tid]; }

  const float xn0 = x[3*n], xn1 = x[3*n+1], xn2 = x[3*n+2];
  float F[9];
  #pragma unroll
  for (int j = 0; j < 9; ++j) F[j] = frames[9*n + j];
  __syncthreads();

  const int brow0 = (half * 2)     * EHDIM + li;  // row K=0 or K=2
  const int brow1 = (half * 2 + 1) * EHDIM + li;  // row K=1 or K=3

  // Each wave handles 128 edges = 8 tiles of 16 edges.
  for (int t = 0; t < 8; ++t) {
    const int base = wave * 128 + t * 16;
    const int m = base + li;
    const float r0 = x[3*m]   - xn0;
    const float r1 = x[3*m+1] - xn1;
    const float r2 = x[3*m+2] - xn2;
    // rel_local[i] = sum_j frames[n,j,i] * rel[j]  (frames^T @ rel)
    const float rl0 = F[0]*r0 + F[3]*r1 + F[6]*r2;
    const float rl1 = F[1]*r0 + F[4]*r1 + F[7]*r2;
    const float rl2 = F[2]*r0 + F[5]*r1 + F[8]*r2;
    const float dist = sqrtf(rl0*rl0 + rl1*rl1 + rl2*rl2);
    if (half == 0) { s_rl0[m] = rl0; s_rl1[m] = rl1; s_rl2[m] = rl2; }

    v2f a;
    a.x = half ? rl2 : rl0;   // K=2 : K=0
    a.y = half ? dist : rl1;  // K=3 : K=1

    float acc[8] = {0.f,0.f,0.f,0.f,0.f,0.f,0.f,0.f};
    #pragma unroll 2
    for (int g = 0; g < 8; ++g) {
      const int gc = g * 16;
      v2f b;
      b.x = s_w1[brow0 + gc];
      b.y = s_w1[brow1 + gc];
      const float hbias = s_b1[gc + li];
      const float hw2   = s_w2[gc + li];
      const v8f zero8 = {0.f,0.f,0.f,0.f,0.f,0.f,0.f,0.f};
      v8f d = __builtin_amdgcn_wmma_f32_16x16x4_f32(false, a, false, b, (short)0, zero8, false, false);
      #pragma unroll
      for (int r = 0; r < 8; ++r)
        acc[r] += silu_f(d[r] + hbias) * hw2;
    }
    // Reduce each edge's 128 hidden contributions across the 16-lane halves.
    #pragma unroll
    for (int r = 0; r < 8; ++r) {
      #pragma unroll
      for (int mk = 1; mk <= 8; mk <<= 1)
        acc[r] += __shfl_xor(acc[r], mk, 32);
    }
    if (lane == 0)
      for (int r = 0; r < 8; ++r) s_logit[base + r] = acc[r];
    if (lane == 16)
      for (int r = 0; r < 8; ++r) s_logit[base + 8 + r] = acc[r];
  }
  __syncthreads();

  // logits[m] = edge_logit + dst[m]  (src[n] and eg2_b cancel in softmax); mask diag.
  float lv[4]; float lmax = -3.0e38f;
  #pragma unroll
  for (int j = 0; j < 4; ++j) {
    const int m = tid + (j << 8);
    float l = s_logit[m] + dstv[m];
    if (m == n) l = -3.0e38f;
    lv[j] = l;
    lmax = fmaxf(lmax, l);
  }
  #pragma unroll
  for (int mk = 16; mk; mk >>= 1) lmax = fmaxf(lmax, __shfl_xor(lmax, mk, 32));
  if (lane == 0) s_red[wave] = lmax;
  __syncthreads();
  float gmax = s_red[0];
  #pragma unroll
  for (int w = 1; w < 8; ++w) gmax = fmaxf(gmax, s_red[w]);
  __syncthreads();

  float se = 0.f, m0s = 0.f, m1s = 0.f, m2s = 0.f;
  #pragma unroll
  for (int j = 0; j < 4; ++j) {
    const int m = tid + (j << 8);
    if (m == n) continue;
    const float e = __expf(lv[j] - gmax);
    se += e;
    const float w = e * nsv[m];
    m0s += w * s_rl0[m];
    m1s += w * s_rl1[m];
    m2s += w * s_rl2[m];
  }
  #pragma unroll
  for (int mk = 16; mk; mk >>= 1) {
    se  += __shfl_xor(se,  mk, 32);
    m0s += __shfl_xor(m0s, mk, 32);
    m1s += __shfl_xor(m1s, mk, 32);
    m2s += __shfl_xor(m2s, mk, 32);
  }
  if (lane == 0) {
    s_red[wave * 4 + 0] = se;
    s_red[wave * 4 + 1] = m0s;
    s_red[wave * 4 + 2] = m1s;
    s_red[wave * 4 + 3] = m2s;
  }
  __syncthreads();
  if (tid == 0) {
    float S = 0.f, a0 = 0.f, a1 = 0.f, a2 = 0.f;
    #pragma unroll
    for (int w = 0; w < 8; ++w) {
      S  += s_red[w * 4 + 0];
      a0 += s_red[w * 4 + 1];
      a1 += s_red[w * 4 + 2];
      a2 += s_red[w * 4 + 3];
    }
    const float inv = 1.0f / fmaxf(S, 1e-20f);
    msg[3*n + 0] = a0 * inv;
    msg[3*n + 1] = a1 * inv;
    msg[3*n + 2] = a2 * inv;
  }
}

// out[n] = hid2[n] @ out2_w + out2_b + 0.25 * msg[n]. One wave per node.
__global__ void out2_kernel(const float* __restrict__ hid2,
                            const float* __restrict__ out2_w,
                            const float* __restrict__ out2_b,
                            const float* __restrict__ msg,
                            float* __restrict__ out) {
  const int wave = threadIdx.x >> 5;
  const int lane = threadIdx.x & 31;
  const int node = blockIdx.x * 8 + wave;
  float p0 = 0.f, p1 = 0.f, p2 = 0.f;
  #pragma unroll
  for (int j = 0; j < 8; ++j) {
    const int k = lane + (j << 5);
    const float v = hid2[node * HDIM + k];
    p0 += v * out2_w[k * 3 + 0];
    p1 += v * out2_w[k * 3 + 1];
    p2 += v * out2_w[k * 3 + 2];
  }
  #pragma unroll
  for (int mk = 16; mk; mk >>= 1) {
    p0 += __shfl_xor(p0, mk, 32);
    p1 += __shfl_xor(p1, mk, 32);
    p2 += __shfl_xor(p2, mk, 32);
  }
  if (lane == 0) {
    out[node * 3 + 0] = p0 + out2_b[0] + 0.25f * msg[node * 3 + 0];
    out[node * 3 + 1] = p1 + out2_b[1] + 0.25f * msg[node * 3 + 1];
    out[node * 3 + 2] = p2 + out2_b[2] + 0.25f * msg[node * 3 + 2];
  }
}

extern "C" void kernel_launch(void* const* d_in, const int* in_sizes, int n_in,
                              void* d_out, int out_size, void* d_ws, size_t ws_size,
                              hipStream_t stream) {
  const float* h      = (const float*)d_in[0];
  const float* delta  = (const float*)d_in[1];
  const float* t_emb  = (const float*)d_in[2];
  const float* x      = (const float*)d_in[3];
  const float* frames = (const float*)d_in[4];
  const float* np1_w  = (const float*)d_in[5];
  const float* np1_b  = (const float*)d_in[6];
  const float* np2_w  = (const float*)d_in[7];
  const float* np2_b  = (const float*)d_in[8];
  // d_in[9]/d_in[10] (src_w/src_b): constant along the softmax axis -> cancels, skipped.
  const float* dst_w  = (const float*)d_in[11];
  const float* dst_b  = (const float*)d_in[12];
  const float* ns_w   = (const float*)d_in[13];
  const float* ns_b   = (const float*)d_in[14];
  const float* eg1_w  = (const float*)d_in[15];
  const float* eg1_b  = (const float*)d_in[16];
  const float* eg2_w  = (const float*)d_in[17];
  // d_in[18] (eg2_b): constant along the softmax axis -> cancels, skipped.
  const float* out1_w = (const float*)d_in[19];
  const float* out1_b = (const float*)d_in[20];
  const float* out2_w = (const float*)d_in[21];
  const float* out2_b = (const float*)d_in[22];

  float* ws   = (float*)d_ws;
  float* hid  = ws;                         // N*H scratch (hid1, later hid2)
  float* nh   = hid  + N_NODES * HDIM;      // N*H node_hidden
  float* dstv = nh   + N_NODES * HDIM;      // N
  float* nsv  = dstv + N_NODES;             // N
  float* msg  = nsv  + N_NODES;             // N*3
  float* outv = (float*)d_out;

  dim3 gtile(HDIM / 16, N_NODES / 16);  // 16 x 64 tiles, one wave each
  // hid1 = silu([h|delta|t_emb] @ np1 + b)
  gemm_bias_silu_wmma<true><<<gtile, 32, 0, stream>>>(h, delta, t_emb,
                                                      np1_w, np1_b, hid, HDIM);
  // node_hidden = silu(hid1 @ np2 + b)
  gemm_bias_silu_wmma<false><<<gtile, 32, 0, stream>>>(hid, nullptr, nullptr,
                                                       np2_w, np2_b, nh, HDIM);
  heads_kernel<<<N_NODES / 8, 256, 0, stream>>>(nh, dst_w, dst_b, ns_w, ns_b, dstv, nsv);
  edge_attn_msg_kernel<<<N_NODES, 256, 0, stream>>>(x, frames, dstv, nsv,
                                                    eg1_w, eg1_b, eg2_w, msg);
  // hid2 = silu([node_hidden|delta|msg] @ out1 + b)
  gemm_bias_silu_wmma<true><<<gtile, 32, 0, stream>>>(nh, delta, msg,
                                                      out1_w, out1_b, hid, HDIM);
  out2_kernel<<<N_NODES / 8, 256, 0, stream>>>(hid, out2_w, out2_b, msg, outv);
}